// Similarity_15281493639174
// MI455X (gfx1250) — compile-verified
//
#include <hip/hip_runtime.h>

typedef __attribute__((ext_vector_type(2))) float    v2f;
typedef __attribute__((ext_vector_type(8))) float    v8f;
typedef __attribute__((ext_vector_type(4))) unsigned v4u;
typedef __attribute__((ext_vector_type(4))) int      v4i;
typedef __attribute__((ext_vector_type(8))) int      v8i;

// Problem geometry (fixed by the reference)
constexpr int kB  = 64;
constexpr int kTc = 1024;
constexpr int kTq = 128;
constexpr int kD  = 512;

// Tiling
constexpr int kKC   = 64;            // K chunk staged in LDS
constexpr int kRows = 128;           // context rows per workgroup
constexpr int kLdsStride = kKC + 2;  // pad: 2 DWORDs per 64 (matches TDM pad cfg)

// ---------------------------------------------------------------------------
// Kernel 1: qterm[b,q] = query[b,q,:] . w_q + bias   (one wave per row)
// ---------------------------------------------------------------------------
__global__ __launch_bounds__(256)
void sim_qterm_kernel(const float* __restrict__ query,
                      const float* __restrict__ kern,
                      const float* __restrict__ bias,
                      float* __restrict__ qterm,
                      int n_rows) {
    const int lane = threadIdx.x & 31;
    const int wave = threadIdx.x >> 5;
    const int row  = blockIdx.x * 8 + wave;
    if (row >= n_rows) return;

    const float* q  = query + (size_t)row * kD;
    const float* wq = kern + kD; // w_q = kernel[d : 2d]

    float s = 0.0f;
    for (int i = lane; i < kD; i += 32)
        s += q[i] * wq[i];

    #pragma unroll
    for (int off = 16; off >= 1; off >>= 1)
        s += __shfl_xor(s, off, 32);

    if (lane == 0) qterm[row] = s + bias[0];
}

// ---------------------------------------------------------------------------
// Kernel 2: out[b,t,q] = sum_k ctx[b,t,k]*(w_m[k]*qry[b,q,k] + w_c[k])
//                        + qterm[b,q]
// A-operand (context) is staged global->LDS by the Tensor Data Mover with
// hardware row padding; B-operand (query) is VALU-fused (w_m*q + w_c) into
// LDS while the TDM runs in the background.
// ---------------------------------------------------------------------------
__global__ __launch_bounds__(256)
void sim_gemm_kernel(const float* __restrict__ context,
                     const float* __restrict__ query,
                     const float* __restrict__ kern,
                     const float* __restrict__ qterm,
                     float* __restrict__ out) {
    __shared__ float As[kRows][kLdsStride];  // raw context chunk (TDM-filled)
    __shared__ float Bs[kTq][kLdsStride];    // w_m * query + w_c

    const int b       = blockIdx.y;
    const int rowBase = blockIdx.x * kRows;
    const int tid     = threadIdx.x;
    const int lane    = tid & 31;
    const int wave    = tid >> 5;

    const float* ctxB = context + ((size_t)b * kTc + rowBase) * kD;
    const float* qryB = query   + (size_t)b * kTq * kD;
    const float* wc   = kern;            // w_c = kernel[0 : d]
    const float* wm   = kern + 2 * kD;   // w_m = kernel[2d : 3d]

    const unsigned ldsA = (unsigned)(uintptr_t)&As[0][0];

    v8f acc[8];
    #pragma unroll
    for (int i = 0; i < 8; ++i)
        acc[i] = (v8f){0.f, 0.f, 0.f, 0.f, 0.f, 0.f, 0.f, 0.f};

    const int m     = lane & 15;          // M (A) / N (B) index of this lane
    const int khalf = (lane >> 4) << 1;   // lanes 16..31 hold K = k0+2, k0+3

    #pragma unroll 1
    for (int kc = 0; kc < kD; kc += kKC) {
        __syncthreads();   // previous chunk fully consumed before overwrite

        // ---- TDM: async-copy 128x64 context tile into As (wave 0 issues) ----
        // D# group0: count=1, lds_addr, 57b global addr, type=2
        // D# group1: data_size=4B, pad_enable (2 DW pad per 64 DW -> stride 66),
        //            tensor_dim0=512, tensor_dim1=128, tile=64x128, stride0=512
        if (wave == 0) {
            const unsigned long long ga =
                (unsigned long long)(uintptr_t)(ctxB + kc);
            v4u g0 = { 1u,                                   // count=1 (valid)
                       ldsA,                                 // lds_addr (bytes)
                       (unsigned)(ga & 0xFFFFFFFFu),         // global_addr[31:0]
                       (unsigned)((ga >> 32) & 0x1FFFFFFu)   // global_addr[56:32]
                           | (2u << 30) };                   // type=2 ("image")
            v8i g1 = { (int)((2u << 16)      // data_size = 4 bytes
                           | (1u << 20)      // pad_enable
                           | (5u << 22)      // pad_interval: 64 DWORDs
                           | (1u << 25)),    // pad_amount: 2 DWORDs
                       (int)((unsigned)kD << 16),     // tensor_dim0[15:0]=512
                       (int)((unsigned)kRows << 16),  // dim0[31:16]=0 | tensor_dim1=128
                       (int)((unsigned)kKC << 16),    // dim1[31:16]=0 | tile_dim0=64
                       (int)(unsigned)kRows,          // tile_dim1=128 | tile_dim2=0
                       (int)(unsigned)kD,             // tensor_dim0_stride=512
                       0, 0 };                        // dim1_stride unused (2D)
            v4i gz = {0, 0, 0, 0};
#if __clang_major__ >= 23
            v8i gz8 = {0, 0, 0, 0, 0, 0, 0, 0};
            __builtin_amdgcn_tensor_load_to_lds(g0, g1, gz, gz, gz8, 0);
#else
            __builtin_amdgcn_tensor_load_to_lds(g0, g1, gz, gz, 0);
#endif
        }

        // ---- stage Bs while the TDM runs: fused w_m * qry + w_c ----
        #pragma unroll
        for (int i = 0; i < 8; ++i) {
            const int idx = tid + i * 256;        // 0..2047 float4 slots
            const int r   = idx >> 4;             // q row
            const int c4  = (idx & 15) << 2;      // k offset (x4 floats)
            const float4 qv  = *(const float4*)(qryB + (size_t)r * kD + kc + c4);
            const float4 wmv = *(const float4*)(wm + kc + c4);
            const float4 wcv = *(const float4*)(wc + kc + c4);
            float* dst = &Bs[r][c4];
            dst[0] = wmv.x * qv.x + wcv.x;
            dst[1] = wmv.y * qv.y + wcv.y;
            dst[2] = wmv.z * qv.z + wcv.z;
            dst[3] = wmv.w * qv.w + wcv.w;
        }

        // ---- prefetch next query chunk ----
        if (kc + kKC < kD) {
            const int pr = tid >> 1;
            const int pc = (tid & 1) << 5;
            __builtin_prefetch(qryB + (size_t)pr * kD + kc + kKC + pc, 0, 0);
        }

        // Issuing wave waits for the TDM; barrier publishes As+Bs to the WG.
        if (wave == 0)
            __builtin_amdgcn_s_wait_tensorcnt(0);
        __syncthreads();

        // ---- WMMA inner loop: K step of 4 ----
        #pragma unroll
        for (int k0 = 0; k0 < kKC; k0 += 4) {
            // A fragment (16x4 f32): lane<16 -> K=k0,k0+1 ; lane>=16 -> k0+2,k0+3
            const v2f a = *(const v2f*)&As[wave * 16 + m][k0 + khalf];
            #pragma unroll
            for (int ct = 0; ct < 8; ++ct) {
                // B fragment (4x16 f32), same K striping, N = lane&15
                const v2f bf = *(const v2f*)&Bs[ct * 16 + m][k0 + khalf];
                acc[ct] = __builtin_amdgcn_wmma_f32_16x16x4_f32(
                    false, a, false, bf, (short)0, acc[ct], false, false);
            }
        }
    }

    // ---- epilogue: add (q_term + bias), store per C/D layout ----
    // C/D 16x16 f32: VGPR j -> M = j + 8*(lane>=16), N = lane&15
    const int hi = lane >> 4;
    float* outB = out + ((size_t)b * kTc + rowBase + wave * 16) * kTq;
    #pragma unroll
    for (int ct = 0; ct < 8; ++ct) {
        const int col = ct * 16 + m;
        const float qv = qterm[b * kTq + col];
        #pragma unroll
        for (int j = 0; j < 8; ++j) {
            const int r = j + 8 * hi;
            outB[(size_t)r * kTq + col] = acc[ct][j] + qv;
        }
    }
}

// ---------------------------------------------------------------------------
extern "C" void kernel_launch(void* const* d_in, const int* in_sizes, int n_in,
                              void* d_out, int out_size, void* d_ws, size_t ws_size,
                              hipStream_t stream) {
    const float* context = (const float*)d_in[0];
    const float* query   = (const float*)d_in[1];
    const float* kern    = (const float*)d_in[2];
    const float* bias    = (const float*)d_in[3];
    float* out   = (float*)d_out;
    float* qterm = (float*)d_ws;   // kB * kTq floats = 32 KB scratch

    const int n_rows = kB * kTq;   // 8192 q-rows
    sim_qterm_kernel<<<dim3(n_rows / 8), dim3(256), 0, stream>>>(
        query, kern, bias, qterm, n_rows);

    sim_gemm_kernel<<<dim3(kTc / kRows, kB), dim3(256), 0, stream>>>(
        context, query, kern, qterm, out);
}